// MultiHeadedDiff_44006234915087
// MI455X (gfx1250) — compile-verified
//
#include <hip/hip_runtime.h>
#include <hip/hip_bf16.h>

#define SS 1024
#define DD 512
#define HH 8
#define DKK 64

typedef __attribute__((ext_vector_type(2))) float v2f;
typedef __attribute__((ext_vector_type(8))) float v8f;
typedef __attribute__((ext_vector_type(4))) unsigned int u32x4;
typedef __attribute__((ext_vector_type(8))) int i32x8;
typedef __attribute__((ext_vector_type(4))) int i32x4;

// SPECIFIC relation ids as a 64-bit bitmask:
// {0,1,3,4,5,6,7,8,12,13,15,16,17,22,23,28,32,33,35,36}
#define SPECMASK 0x0000001B10C3B1FBull

__device__ __forceinline__ v8f wmma4(v2f a, v2f b, v8f c) {
  // V_WMMA_F32_16X16X4_F32: D = A(16x4 f32) * B(4x16 f32) + C(16x16 f32)
  return __builtin_amdgcn_wmma_f32_16x16x4_f32(
      false, a, false, b, (short)0, c, false, false);
}

// ---------------------------------------------------------------------------
// TDM: async DMA of a 2D tile [64 rows x 64 f32] from global to LDS, with
// 4-DWORD padding after every 64 DWORDs -> LDS row stride 68 (bank-conflict-
// free B-fragment reads). Descriptor per CDNA5 ISA.
// 6-arg builtin form (clang-23 / therock headers).
// ---------------------------------------------------------------------------
__device__ __forceinline__ void tdm_load_tile(unsigned lds_byte_addr,
                                              const void* gptr) {
  unsigned long long ga = (unsigned long long)(size_t)gptr;
  u32x4 g0;
  g0[0] = 1u;                                         // count=1 (valid), load
  g0[1] = lds_byte_addr;                              // lds_addr (bytes)
  g0[2] = (unsigned)(ga & 0xFFFFFFFFull);             // global_addr[31:0]
  g0[3] = (unsigned)((ga >> 32) & 0x01FFFFFFull) | (2u << 30);  // [56:32]+type=2
  i32x8 g1;
  // data_size=2 (4B) | pad_enable | pad_interval=5 (64 DW) | pad_amount=3 (4 DW)
  g1[0] = (2 << 16) | (1 << 20) | (5 << 22) | (3 << 25);
  g1[1] = (64 << 16);          // tensor_dim0 = 64 (low 16 bits in [63:48])
  g1[2] = 0;                   // tensor_dim0 hi = 0, tensor_dim1 lo16 = 0
  g1[3] = 16 | (64 << 16);     // tensor_dim1 = 1<<20 rows, tile_dim0 = 64
  g1[4] = 64;                  // tile_dim1 = 64, tile_dim2 = 0
  g1[5] = 64;                  // tensor_dim0_stride = 64 (low 32 of 48)
  g1[6] = 0;
  g1[7] = 0;
  i32x4 z4 = {0, 0, 0, 0};     // groups 2/3 unused for 2D tensors
  i32x8 z8 = {0, 0, 0, 0, 0, 0, 0, 0};
  __builtin_amdgcn_tensor_load_to_lds(g0, g1, z4, z4, z8, 0);
}

__device__ __forceinline__ float waveMax(float v) {
  for (int o = 16; o; o >>= 1) v = fmaxf(v, __shfl_xor(v, o));
  return v;
}
__device__ __forceinline__ float waveSum(float v) {
  for (int o = 16; o; o >>= 1) v += __shfl_xor(v, o);
  return v;
}
__device__ __forceinline__ void ins3(float v, int i,
                                     float& v0, int& i0, float& v1, int& i1,
                                     float& v2, int& i2) {
  if (v > v0)      { v2 = v1; i2 = i1; v1 = v0; i1 = i0; v0 = v; i0 = i; }
  else if (v > v1) { v2 = v1; i2 = i1; v1 = v;  i1 = i; }
  else if (v > v2) { v2 = v;  i2 = i; }
}

// ---------------------------------------------------------------------------
// Out[M,N] = (X[M,K] @ W[N,K]^T + bias[N]) * scale ; one wave per 16x16 tile.
// ---------------------------------------------------------------------------
__global__ void gemm_xwt(const float* __restrict__ X, const float* __restrict__ W,
                         const float* __restrict__ bias, float* __restrict__ Out,
                         int N, int Kdim, float scale) {
  const int tiles_n = N >> 4;
  const int i0 = (blockIdx.x / tiles_n) << 4;
  const int j0 = (blockIdx.x % tiles_n) << 4;
  const int lane = threadIdx.x & 31;
  const int mr = lane & 15;
  const int kb = (lane >> 4) << 1;
  v8f acc = {};
  for (int k = 0; k < Kdim; k += 4) {
    v2f a, b;
    a.x = X[(size_t)(i0 + mr) * Kdim + k + kb];
    a.y = X[(size_t)(i0 + mr) * Kdim + k + kb + 1];
    b.x = W[(size_t)(j0 + mr) * Kdim + k + kb];       // B[k'][n] = W[j0+n][k']
    b.y = W[(size_t)(j0 + mr) * Kdim + k + kb + 1];
    acc = wmma4(a, b, acc);
  }
  const int col = lane & 15, rowb = (lane >> 4) << 3;
  for (int r = 0; r < 8; ++r) {
    const int row = i0 + rowb + r, cj = j0 + col;
    Out[(size_t)row * N + cj] = (acc[r] + bias[cj]) * scale;
  }
}

// ---------------------------------------------------------------------------
// logits[h,q,k] = Qs[q, h*64+d] . Kb[k, h*64+d]   (Qs pre-scaled by 1/8)
// ---------------------------------------------------------------------------
__global__ void qk_logits(const float* __restrict__ Qs, const float* __restrict__ Kb,
                          float* __restrict__ logits) {
  const int q0 = blockIdx.x << 4, k0 = blockIdx.y << 4;
  const int lane = threadIdx.x & 31;
  const int mr = lane & 15, kb = (lane >> 4) << 1;
  const int col = lane & 15, rowb = (lane >> 4) << 3;
  for (int h = 0; h < HH; ++h) {
    v8f acc = {};
    const float* qh = Qs + h * DKK;
    const float* kh = Kb + h * DKK;
    for (int d = 0; d < DKK; d += 4) {
      v2f a, b;
      a.x = qh[(size_t)(q0 + mr) * DD + d + kb];
      a.y = qh[(size_t)(q0 + mr) * DD + d + kb + 1];
      b.x = kh[(size_t)(k0 + mr) * DD + d + kb];
      b.y = kh[(size_t)(k0 + mr) * DD + d + kb + 1];
      acc = wmma4(a, b, acc);
    }
    float* lp = logits + ((size_t)h * SS + q0) * SS + k0;
    for (int r = 0; r < 8; ++r)
      lp[(size_t)(rowb + r) * SS + col] = acc[r];
  }
}

// ---------------------------------------------------------------------------
// logits[h,q,k] += Qs[q, h*64+d] . relk[q,k,d] ; per-q batched (heads padded
// to M=16). relation_k streamed via double-buffered TDM into padded LDS.
// One block per q (4 waves), 16 kk-blocks of 64.
// ---------------------------------------------------------------------------
__global__ void __launch_bounds__(128) rel_logits(
    const float* __restrict__ Qs, const float* __restrict__ relk,
    float* __restrict__ logits) {
  __shared__ float lds[2][64 * 68];
  const int q = blockIdx.x;
  const float* base = relk + (size_t)q * SS * DKK;
  const unsigned lds0 = (unsigned)(size_t)(void*)&lds[0][0];
  const unsigned lds1 = (unsigned)(size_t)(void*)&lds[1][0];
  const int tid = threadIdx.x;
  const int wave = tid >> 5;
  const int lane = tid & 31;
  const int mr = lane & 15, kb = (lane >> 4) << 1;
  const int col = lane & 15, rowb = (lane >> 4) << 3;

  if (tid < 32) {                          // wave 0 drives the TDM pipeline
    tdm_load_tile(lds0, base);             // kk-block 0 -> buffer 0
  }
  for (int blk = 0; blk < 16; ++blk) {
    if (tid < 32) {
      if (blk + 1 < 16) {
        tdm_load_tile(((blk + 1) & 1) ? lds1 : lds0,
                      base + (size_t)(blk + 1) * 64 * DKK);
        __builtin_amdgcn_s_wait_tensorcnt(1);   // current tile done, next in flight
      } else {
        __builtin_amdgcn_s_wait_tensorcnt(0);
      }
    }
    __syncthreads();                       // publish tile to all 4 waves
    const float* buf = (blk & 1) ? &lds[1][0] : &lds[0][0];
    v8f acc = {};
    for (int d = 0; d < DKK; d += 4) {
      v2f a, b;
      float a0 = 0.f, a1 = 0.f;
      if (mr < HH) {                       // rows 8..15 are zero padding
        a0 = Qs[(size_t)q * DD + mr * DKK + d + kb];
        a1 = Qs[(size_t)q * DD + mr * DKK + d + kb + 1];
      }
      a.x = a0; a.y = a1;
      b.x = buf[((wave << 4) + mr) * 68 + d + kb];
      b.y = buf[((wave << 4) + mr) * 68 + d + kb + 1];
      acc = wmma4(a, b, acc);
    }
    const int kt = (blk << 6) + (wave << 4);
    for (int r = 0; r < 8; ++r) {
      const int hh = rowb + r;
      if (hh < HH) {
        const size_t off = ((size_t)hh * SS + q) * SS + kt + col;
        logits[off] += acc[r];
      }
    }
    __syncthreads();                       // done with buf before TDM reuses it
  }
}

// ---------------------------------------------------------------------------
// col[q,k] = edge ? sum_h logits[h,q,k] : -inf
// ---------------------------------------------------------------------------
__global__ void colsum(const float* __restrict__ logits, const int* __restrict__ relation,
                       float* __restrict__ col) {
  const int idx = blockIdx.x * blockDim.x + threadIdx.x;
  if (idx >= SS * SS) return;
  const int rel = relation[idx];
  const bool spec = (rel >= 0) && (rel < 64) && ((SPECMASK >> rel) & 1ull);
  float s = 0.f;
  for (int h = 0; h < HH; ++h) s += logits[((size_t)h << 20) + idx];
  col[idx] = spec ? -__builtin_inff() : s;
}

// ---------------------------------------------------------------------------
// Per column k: top-3 rows by col[], plus per-head norm over the kept set.
// One wave per column; per-lane insertion top-3 + shfl merge.
// ---------------------------------------------------------------------------
__global__ void topk3(const float* __restrict__ col, const float* __restrict__ logits,
                      int* __restrict__ topidx, float* __restrict__ norm) {
  const int kk = blockIdx.x;
  const int lane = threadIdx.x;
  const float NEG = -__builtin_inff();
  float v0 = NEG, v1 = NEG, v2 = NEG;
  int i0 = -1, i1 = -1, i2 = -1;
  for (int q = lane; q < SS; q += 32)
    ins3(col[(size_t)q * SS + kk], q, v0, i0, v1, i1, v2, i2);
  for (int off = 16; off; off >>= 1) {
    float w0 = __shfl_down(v0, off), w1 = __shfl_down(v1, off), w2 = __shfl_down(v2, off);
    int   j0 = __shfl_down(i0, off), j1 = __shfl_down(i1, off), j2 = __shfl_down(i2, off);
    ins3(w0, j0, v0, i0, v1, i1, v2, i2);
    ins3(w1, j1, v0, i0, v1, i1, v2, i2);
    ins3(w2, j2, v0, i0, v1, i1, v2, i2);
  }
  if (lane == 0) {
    if (v0 == NEG) i0 = -1;
    if (v1 == NEG) i1 = -1;
    if (v2 == NEG) i2 = -1;
    topidx[kk * 3 + 0] = i0;
    topidx[kk * 3 + 1] = i1;
    topidx[kk * 3 + 2] = i2;
    for (int h = 0; h < HH; ++h) {
      float s = 0.f;
      if (i0 >= 0) s += logits[((size_t)h * SS + i0) * SS + kk];
      if (i1 >= 0) s += logits[((size_t)h * SS + i1) * SS + kk];
      if (i2 >= 0) s += logits[((size_t)h * SS + i2) * SS + kk];
      norm[h * SS + kk] = s;
    }
  }
}

// ---------------------------------------------------------------------------
// In-place over logits: diffusion scores then row softmax -> P. Block=(h,q).
// ---------------------------------------------------------------------------
__global__ void scores_softmax(float* __restrict__ P, const int* __restrict__ relation,
                               const int* __restrict__ topidx,
                               const float* __restrict__ norm) {
  const int hq = blockIdx.x;
  const int h = hq >> 10, q = hq & 1023;
  float* row = P + ((size_t)h * SS + q) * SS;
  const int* relrow = relation + (size_t)q * SS;
  __shared__ float redm[8], reds[8];
  const float NEG = -__builtin_inff();
  float vals[4];
  float lmax = NEG;
  for (int t = 0; t < 4; ++t) {
    const int kk = threadIdx.x + (t << 8);
    const float s = row[kk];
    const int rel = relrow[kk];
    const bool spec = (rel >= 0) && (rel < 64) && ((SPECMASK >> rel) & 1ull);
    float o;
    if (spec) {
      o = s;
    } else {
      const int* ti = topidx + kk * 3;
      const bool keep = (q == ti[0]) | (q == ti[1]) | (q == ti[2]);
      o = keep ? s / norm[h * SS + kk] : 0.f;
    }
    vals[t] = o;
    lmax = fmaxf(lmax, o);
  }
  const int wid = threadIdx.x >> 5, lid = threadIdx.x & 31;
  float m = waveMax(lmax);
  if (lid == 0) redm[wid] = m;
  __syncthreads();
  float bmax = redm[0];
  for (int i = 1; i < 8; ++i) bmax = fmaxf(bmax, redm[i]);
  float lsum = 0.f;
  for (int t = 0; t < 4; ++t) {
    vals[t] = __expf(vals[t] - bmax);
    lsum += vals[t];
  }
  float sm = waveSum(lsum);
  if (lid == 0) reds[wid] = sm;
  __syncthreads();
  float bsum = 0.f;
  for (int i = 0; i < 8; ++i) bsum += reds[i];
  const float inv = 1.0f / bsum;
  for (int t = 0; t < 4; ++t) {
    const int kk = threadIdx.x + (t << 8);
    row[kk] = vals[t] * inv;
  }
}

// ---------------------------------------------------------------------------
// X[q, h*64+d] = sum_k P[h,q,k] * relv[q,k,d] ; per-q batched, heads padded.
// 4 waves per block, one per 16-wide d subtile; streams relation_v once,
// with global_prefetch hints ahead of the read cursor.
// ---------------------------------------------------------------------------
__global__ void __launch_bounds__(128) rel_values(
    const float* __restrict__ P, const float* __restrict__ relv,
    float* __restrict__ X) {
  const int q = blockIdx.x;
  const int tid = threadIdx.x;
  const int wave = tid >> 5, lane = tid & 31;
  const int mr = lane & 15, kb = (lane >> 4) << 1;
  const int d0 = wave << 4;
  const float* rv = relv + (size_t)q * SS * DKK;
  v8f acc = {};
  for (int kk = 0; kk < SS; kk += 4) {
    if ((kk & 63) == 0 && kk + 64 < SS)
      __builtin_prefetch(rv + (size_t)(kk + 64) * DKK + d0 + mr, 0, 0);
    v2f a, b;
    float a0 = 0.f, a1 = 0.f;
    if (mr < HH) {
      a0 = P[((size_t)mr * SS + q) * SS + kk + kb];
      a1 = P[((size_t)mr * SS + q) * SS + kk + kb + 1];
    }
    a.x = a0; a.y = a1;
    b.x = rv[(size_t)(kk + kb) * DKK + d0 + mr];
    b.y = rv[(size_t)(kk + kb + 1) * DKK + d0 + mr];
    acc = wmma4(a, b, acc);
  }
  const int col = lane & 15, rowb = (lane >> 4) << 3;
  for (int r = 0; r < 8; ++r) {
    const int hh = rowb + r;
    if (hh < HH)
      X[(size_t)q * DD + hh * DKK + d0 + col] = acc[r];
  }
}

// ---------------------------------------------------------------------------
// X[q, h*64+d] += sum_k P[h,q,k] * V[h,k,d]  (per-head GEMM)
// ---------------------------------------------------------------------------
__global__ void pv_values(const float* __restrict__ P, const float* __restrict__ Vb,
                          float* __restrict__ X) {
  const int q0 = blockIdx.x << 4;
  const int d0 = blockIdx.y << 4;
  const int h = blockIdx.z;
  const int lane = threadIdx.x & 31;
  const int mr = lane & 15, kb = (lane >> 4) << 1;
  const float* Ph = P + (size_t)h * SS * SS;
  const float* vh = Vb + h * DKK;
  v8f acc = {};
  for (int kk = 0; kk < SS; kk += 4) {
    v2f a, b;
    a.x = Ph[(size_t)(q0 + mr) * SS + kk + kb];
    a.y = Ph[(size_t)(q0 + mr) * SS + kk + kb + 1];
    b.x = vh[(size_t)(kk + kb) * DD + d0 + mr];
    b.y = vh[(size_t)(kk + kb + 1) * DD + d0 + mr];
    acc = wmma4(a, b, acc);
  }
  const int col = lane & 15, rowb = (lane >> 4) << 3;
  for (int r = 0; r < 8; ++r) {
    const size_t off = (size_t)(q0 + rowb + r) * DD + h * DKK + d0 + col;
    X[off] += acc[r];
  }
}

extern "C" void kernel_launch(void* const* d_in, const int* in_sizes, int n_in,
                              void* d_out, int out_size, void* d_ws, size_t ws_size,
                              hipStream_t stream) {
  const float* query = (const float*)d_in[0];
  const float* key   = (const float*)d_in[1];
  const float* value = (const float*)d_in[2];
  const float* relk  = (const float*)d_in[3];
  const float* relv  = (const float*)d_in[4];
  const int*   relation = (const int*)d_in[5];
  const float* Wq = (const float*)d_in[6];  const float* bq = (const float*)d_in[7];
  const float* Wk = (const float*)d_in[8];  const float* bk = (const float*)d_in[9];
  const float* Wv = (const float*)d_in[10]; const float* bv = (const float*)d_in[11];
  const float* Wo = (const float*)d_in[12]; const float* bo = (const float*)d_in[13];

  char* ws = (char*)d_ws;
  float* Qs     = (float*)(ws);                              // [S,512], pre-scaled 1/8
  float* Kb     = (float*)(ws + (size_t)(2 << 20));          // [S,512]
  float* Vb     = (float*)(ws + (size_t)(4 << 20));          // [S,512]
  float* logits = (float*)(ws + (size_t)(6 << 20));          // [H,S,S] (becomes P)
  float* col    = (float*)(ws + (size_t)(38 << 20));         // [S,S]
  int*   topidx = (int*)  (ws + (size_t)(42 << 20));         // [S,3]
  float* norm   = (float*)(ws + (size_t)(42 << 20) + (64 << 10)); // [H,S]
  float* X      = (float*)(ws + (size_t)(43 << 20));         // [S,512]
  float* out = (float*)d_out;

  // Projections (Q scaled by 1/sqrt(DK)=0.125)
  gemm_xwt<<<2048, 32, 0, stream>>>(query, Wq, bq, Qs, DD, DD, 0.125f);
  gemm_xwt<<<2048, 32, 0, stream>>>(key,   Wk, bk, Kb, DD, DD, 1.0f);
  gemm_xwt<<<2048, 32, 0, stream>>>(value, Wv, bv, Vb, DD, DD, 1.0f);

  // logits = QK^T (+ Q . relation_k^T), relation_k streamed once via TDM
  qk_logits<<<dim3(SS / 16, SS / 16), 32, 0, stream>>>(Qs, Kb, logits);
  rel_logits<<<SS, 128, 0, stream>>>(Qs, relk, logits);

  // diffusion: masked head-sum, per-column top-3, per-head norm
  colsum<<<(SS * SS) / 256, 256, 0, stream>>>(logits, relation, col);
  topk3<<<SS, 32, 0, stream>>>(col, logits, topidx, norm);

  // scores + softmax in place -> P
  scores_softmax<<<HH * SS, 256, 0, stream>>>(logits, relation, topidx, norm);

  // out = P.V + P.relation_v (relation_v streamed once)
  rel_values<<<SS, 128, 0, stream>>>(logits, relv, X);
  pv_values<<<dim3(SS / 16, DKK / 16, HH), 32, 0, stream>>>(logits, Vb, X);

  // final projection
  gemm_xwt<<<2048, 32, 0, stream>>>(X, Wo, bo, out, DD, DD, 1.0f);
}